// PostprocessWithSampling_13640816132357
// MI455X (gfx1250) — compile-verified
//
#include <hip/hip_runtime.h>
#include <cstdint>

// 128-bit vector of 4 dwords -> lowers to global_load_b128 / global_store_b128
typedef int v4i __attribute__((ext_vector_type(4)));

// ---------------------------------------------------------------------------
// Kernel 1: segmented bulk copy, 16B per lane per iteration.
// Segments (in v4i units): 0=attention_mask, 1=generated_tokens,
// 2=generated_tokens_streaming, 3=token_count (largest, ~4.1M vecs).
// Pure bandwidth: one int4 load + one int4 store per lane.
// ---------------------------------------------------------------------------
__global__ __launch_bounds__(256)
void seg_copy_b128(const v4i* __restrict__ s0, v4i* __restrict__ d0, long n0,
                   const v4i* __restrict__ s1, v4i* __restrict__ d1, long n1,
                   const v4i* __restrict__ s2, v4i* __restrict__ d2, long n2,
                   const v4i* __restrict__ s3, v4i* __restrict__ d3, long n3)
{
    const long total  = n0 + n1 + n2 + n3;
    const long stride = (long)gridDim.x * blockDim.x;
    for (long i = (long)blockIdx.x * blockDim.x + threadIdx.x; i < total; i += stride) {
        long j = i;
        if (j < n0) { d0[j] = s0[j]; continue; }
        j -= n0;
        if (j < n1) { d1[j] = s1[j]; continue; }
        j -= n1;
        if (j < n2) { d2[j] = s2[j]; continue; }
        j -= n2;
        // largest segment: token_count. Prefetch ahead (gfx1250 global_prefetch_b8;
        // speculative, OOB addresses are silently dropped).
        __builtin_prefetch(&s3[j + 2048], 0, 0);
        d3[j] = s3[j];
    }
}

// Scalar tail for segment sizes not divisible by 4 (defensive; with the given
// shapes all segment word counts are multiples of 4 so this does ~nothing).
__global__ void seg_copy_tail(const int* __restrict__ s0, int* __restrict__ d0, long b0, long n0,
                              const int* __restrict__ s1, int* __restrict__ d1, long b1, long n1,
                              const int* __restrict__ s2, int* __restrict__ d2, long b2, long n2,
                              const int* __restrict__ s3, int* __restrict__ d3, long b3, long n3)
{
    int t = threadIdx.x;           // one small block handles all tails (<16 words)
    if (t < n0) d0[b0 + t] = s0[b0 + t];
    if (t < n1) d1[b1 + t] = s1[b1 + t];
    if (t < n2) d2[b2 + t] = s2[b2 + t];
    if (t < n3) d3[b3 + t] = s3[b3 + t];
}

// ---------------------------------------------------------------------------
// Kernel 2: per-batch scalar logic + scatters (runs after the copy on the
// same stream, so its point-writes land on top of the copied data).
// B=128 -> one block, 4 wave32s. All scatter targets are distinct per b.
// ---------------------------------------------------------------------------
__global__ void postprocess_scatter(const int* __restrict__ tokens,
                                    const int* __restrict__ lti_in,
                                    const int* __restrict__ gi_in,
                                    const int* __restrict__ tc_in,
                                    int*   __restrict__ out_tokens,
                                    int*   __restrict__ out_lti,
                                    float* __restrict__ out_mask,
                                    int*   __restrict__ out_gt,
                                    int*   __restrict__ out_gts,
                                    int*   __restrict__ out_gi,
                                    int*   __restrict__ out_tc,
                                    int B, int S, int V)
{
    int b = blockIdx.x * blockDim.x + threadIdx.x;
    if (b >= B) return;

    int tok    = tokens[b];
    int lti    = lti_in[b] + 1;  if (lti > S - 1) lti = S - 1;   // clamp
    int gi_old = gi_in[b];
    int gi     = gi_old + 1;     if (gi  > S - 1) gi  = S - 1;   // clamp

    // small outputs
    out_tokens[b] = tok;
    out_lti[b]    = lti;
    out_gi[b]     = gi;

    // scatters into the already-copied big buffers
    out_mask[(long)b * S + lti]   = 1.0f;   // attention_mask[b, lti] = 1
    out_gt [(long)b * S + gi_old] = tok;    // generated_tokens[b, gi_old] = tok
    out_gts[(long)b * S + gi_old] = tok;    // streaming buffer, same
    long tci = (long)b * V + tok;           // histogram bump (row-unique per b)
    out_tc[tci] = tc_in[tci] + 1;
}

// ---------------------------------------------------------------------------
extern "C" void kernel_launch(void* const* d_in, const int* in_sizes, int n_in,
                              void* d_out, int out_size, void* d_ws, size_t ws_size,
                              hipStream_t stream)
{
    // Inputs (setup_inputs order)
    const int* tokens = (const int*)d_in[0];            // (B,)
    const int* lti_in = (const int*)d_in[1];            // (B,1)
    const int* mask_i = (const int*)d_in[2];            // (B,S,1) f32, copied bitwise
    const int* gt_i   = (const int*)d_in[3];            // (B,S,1) i32
    const int* gts_i  = (const int*)d_in[4];            // (B,S,1) i32
    const int* gi_in  = (const int*)d_in[5];            // (B,1)
    const int* tc_i   = (const int*)d_in[6];            // (B,V,1) i32

    const int  B  = in_sizes[0];
    const long BS = (long)in_sizes[2];
    const long BV = (long)in_sizes[6];
    const int  S  = (int)(BS / B);
    const int  V  = (int)(BV / B);

    // Output tuple layout (flat words, return order)
    int* out = (int*)d_out;
    long o_tokens = 0;
    long o_lti    = o_tokens + B;
    long o_mask   = o_lti + B;
    long o_gt     = o_mask + BS;
    long o_gts    = o_gt + BS;
    long o_gi     = o_gts + BS;
    long o_tc     = o_gi + B;

    int*   out_tokens = out + o_tokens;
    int*   out_lti    = out + o_lti;
    float* out_mask   = (float*)(out + o_mask);
    int*   out_gt     = out + o_gt;
    int*   out_gts    = out + o_gts;
    int*   out_gi     = out + o_gi;
    int*   out_tc     = out + o_tc;

    // --- Bulk copy: vectorized 16B main body -------------------------------
    long v0 = BS >> 2, v1 = BS >> 2, v2 = BS >> 2, v3 = BV >> 2;
    long totalVec = v0 + v1 + v2 + v3;
    int  threads  = 256;
    long blocksL  = (totalVec + threads - 1) / threads;
    unsigned blocks = (unsigned)(blocksL > 0x7FFFFFF0L ? 0x7FFFFFF0L : blocksL);
    if (blocks == 0) blocks = 1;

    seg_copy_b128<<<blocks, threads, 0, stream>>>(
        (const v4i*)mask_i, (v4i*)(out + o_mask), v0,
        (const v4i*)gt_i,   (v4i*)(out + o_gt),   v1,
        (const v4i*)gts_i,  (v4i*)(out + o_gts),  v2,
        (const v4i*)tc_i,   (v4i*)(out + o_tc),   v3);

    // --- Scalar tails (word counts not divisible by 4) ---------------------
    long t0 = BS & 3, t1 = BS & 3, t2 = BS & 3, t3 = BV & 3;
    if (t0 | t1 | t2 | t3) {
        seg_copy_tail<<<1, 32, 0, stream>>>(
            mask_i, out + o_mask, BS - t0, t0,
            gt_i,   out + o_gt,   BS - t1, t1,
            gts_i,  out + o_gts,  BS - t2, t2,
            tc_i,   out + o_tc,   BV - t3, t3);
    }

    // --- Per-batch logic + scatters (after copies, same stream) ------------
    int sthreads = 128;
    int sblocks  = (B + sthreads - 1) / sthreads;
    postprocess_scatter<<<sblocks, sthreads, 0, stream>>>(
        tokens, lti_in, gi_in, tc_i,
        out_tokens, out_lti, out_mask, out_gt, out_gts, out_gi, out_tc,
        B, S, V);
}